// Upsample2d_5557687681319
// MI455X (gfx1250) — compile-verified
//
#include <hip/hip_runtime.h>
#include <hip/hip_bf16.h>
#include <math.h>
#include <stdint.h>

// upfirdn2d 2x upsample, 4x4 FIR (StyleGAN2 style), fp32.
// Roofline: 128 MiB read + 512 MiB write -> ~27us floor at 23.3 TB/s; ~1 GFLOP
// of math => purely bandwidth-bound, WMMA intentionally not used.
// CDNA5 path: async global->LDS staging (ASYNCcnt) + NT b128 store stream.

typedef __attribute__((ext_vector_type(4))) float v4f;  // native vector for NT stores

#define TIN_X 32              // input cols per block
#define TIN_Y 16              // input rows per block
#define TW    (TIN_X + 2)     // 34 (tile width incl. halo)
#define TH    (TIN_Y + 2)     // 18 (tile height incl. halo)
#define TSTRIDE 35            // odd stride -> fewer LDS bank conflicts
#define NELEM (TH * TW)       // 612 staged elements

__global__ __launch_bounds__(256)
void upsample2x_fir4(const float* __restrict__ x,
                     const float* __restrict__ k4,   // 4x4 kernel, row-major
                     float* __restrict__ out,
                     int H, int W, int OH, int OW)
{
    __shared__ float tile[TH * TSTRIDE];

    const int tid  = threadIdx.x;
    const int z    = blockIdx.z;
    const int row0 = blockIdx.y * TIN_Y;
    const int col0 = blockIdx.x * TIN_X;

    const float* __restrict__ xz = x + (size_t)z * H * W;

    // ---------- stage input tile (with halo of 1) into LDS ----------
#if defined(__gfx1250__)
    // CDNA5 async global->LDS copy: per-lane global_load_async_to_lds_b32,
    // tracked with ASYNCcnt. OOB halo lanes zero-fill LDS via a normal DS store.
    for (int l = tid; l < NELEM; l += 256) {
        const int r  = l / TW;
        const int c  = l - r * TW;
        const int gy = row0 - 1 + r;
        const int gx = col0 - 1 + c;
        const unsigned ldsOff =
            (unsigned)(uintptr_t)(&tile[r * TSTRIDE + c]); // low 32b = LDS byte offset
        if ((unsigned)gy < (unsigned)H && (unsigned)gx < (unsigned)W) {
            const unsigned gOff = (unsigned)((gy * W + gx) * 4);
            asm volatile("global_load_async_to_lds_b32 %0, %1, %2"
                         :: "v"(ldsOff), "v"(gOff), "s"(xz)
                         : "memory");
        } else {
            tile[r * TSTRIDE + c] = 0.0f;
        }
    }
    asm volatile("s_wait_asynccnt 0x0" ::: "memory");
#else
    for (int l = tid; l < NELEM; l += 256) {
        const int r  = l / TW;
        const int c  = l - r * TW;
        const int gy = row0 - 1 + r;
        const int gx = col0 - 1 + c;
        float v = 0.0f;
        if ((unsigned)gy < (unsigned)H && (unsigned)gx < (unsigned)W)
            v = __builtin_nontemporal_load(&xz[gy * W + gx]);
        tile[r * TSTRIDE + c] = v;
    }
#endif
    __syncthreads();

    // ---------- per-thread compute: 1 input row x 2 input cols -> 2x4 outputs --
    // Uniform 4x4 taps (scalar loads).
    float kv[16];
    #pragma unroll
    for (int i = 0; i < 16; ++i) kv[i] = k4[i];

    const int tx = tid & 15;   // input col-pair index (j = col0 + 2*tx + s)
    const int ty = tid >> 4;   // input row index     (i = row0 + ty)
    const int i  = row0 + ty;
    const int jb = col0 + 2 * tx;

    // 3 rows x 4 cols neighborhood from LDS: x rows {i-1,i,i+1}, cols {jb-1..jb+2}
    float v[3][4];
    #pragma unroll
    for (int r = 0; r < 3; ++r)
        #pragma unroll
        for (int c = 0; c < 4; ++c)
            v[r][c] = tile[(ty + r) * TSTRIDE + (2 * tx + c)];

    // out(2i+dy, 2j+dx): even out-rows use k-rows {0,2} on x rows {i-1,i};
    // odd out-rows use k-rows {1,3} on x rows {i,i+1}. Same structure for cols.
    float e[4], o[4];
    #pragma unroll
    for (int s = 0; s < 2; ++s) {
        e[2*s+0] = kv[0]*v[0][s]   + kv[2]*v[0][s+1] + kv[8]*v[1][s]    + kv[10]*v[1][s+1];
        e[2*s+1] = kv[1]*v[0][s+1] + kv[3]*v[0][s+2] + kv[9]*v[1][s+1]  + kv[11]*v[1][s+2];
        o[2*s+0] = kv[4]*v[1][s]   + kv[6]*v[1][s+1] + kv[12]*v[2][s]   + kv[14]*v[2][s+1];
        o[2*s+1] = kv[5]*v[1][s+1] + kv[7]*v[1][s+2] + kv[13]*v[2][s+1] + kv[15]*v[2][s+2];
    }

    float* __restrict__ oz = out + (size_t)z * OH * OW;
    const int oy = 2 * i;
    const int ox = 2 * jb;      // multiple of 4 -> 16B-aligned b128 stores

    if (oy < OH) {
        if (ox + 3 < OW) {
            v4f ev = { e[0], e[1], e[2], e[3] };
            __builtin_nontemporal_store(ev, (v4f*)(oz + (size_t)oy * OW + ox));
        } else {
            #pragma unroll
            for (int c = 0; c < 4; ++c)
                if (ox + c < OW) oz[(size_t)oy * OW + ox + c] = e[c];
        }
    }
    if (oy + 1 < OH) {
        if (ox + 3 < OW) {
            v4f ov = { o[0], o[1], o[2], o[3] };
            __builtin_nontemporal_store(ov, (v4f*)(oz + (size_t)(oy + 1) * OW + ox));
        } else {
            #pragma unroll
            for (int c = 0; c < 4; ++c)
                if (ox + c < OW) oz[(size_t)(oy + 1) * OW + ox + c] = o[c];
        }
    }
}

extern "C" void kernel_launch(void* const* d_in, const int* in_sizes, int n_in,
                              void* d_out, int out_size, void* d_ws, size_t ws_size,
                              hipStream_t stream) {
    (void)n_in; (void)d_ws; (void)ws_size;
    const float* x  = (const float*)d_in[0];   // (8,256,128,128) fp32
    const float* k4 = (const float*)d_in[1];   // (1,1,4,4) fp32
    float* out = (float*)d_out;

    const int H = 128, W = 128;
    const int NC = in_sizes[0] / (H * W);      // 2048

    // Derive output spatial dims from out_size (expected 2H x 2W = 256x256).
    int osp = out_size / NC;                   // OH*OW
    int d   = (int)(sqrtf((float)osp) + 0.5f);
    const int OH = d, OW = d;

    dim3 grid(W / TIN_X, H / TIN_Y, NC);       // (4, 8, 2048)
    upsample2x_fir4<<<grid, 256, 0, stream>>>(x, k4, out, H, W, OH, OW);
}